// PatchNet_58188216926937
// MI455X (gfx1250) — compile-verified
//
#include <hip/hip_runtime.h>

// MI455X / gfx1250: wave32, WMMA f32 16x16x4 (full f32 precision, K=729 reduction).
// Round 4: iterate only valid h bodies (hlo..hhi) so the row base is always a
// valid image row. Then:
//  - middle batches (1..30): NO address select -- per-slot byte immediates off a
//    single 32-bit voffset (scale_offset GVS addressing), only a value cndmask
//    against the hoisted loop-invariant w-predicate lane-masks. Stray masked
//    reads are within +-144B of the slice => inside the neighboring batch.
//  - edge batches (0, 31): clamp the index with min/max (one v_med3_i32), still
//    no i64 select chain.

typedef __attribute__((ext_vector_type(2))) float v2f;
typedef __attribute__((ext_vector_type(8))) float v8f;

#define GRID_HW 27
#define WPAD    28                    // padded w extent (zero column at w=27)
#define KDIM    10                    // indicator maps (M dim, padded to 16)
#define PATCH   32
#define IMG     224
#define CHN     3
#define KPAD    (GRID_HW * WPAD)      // 756 padded K columns
#define NSLOT   7                     // K=4 chunks per body (28 = 7*4)
#define ROWELEM (IMG * CHN)           // 672 floats per image row
#define SLICE   (IMG * IMG * CHN)     // 150528 floats per batch image
#define OUTK    (PATCH * PATCH * CHN) // 3072 floats per (b,k) slice

struct TrueT  { static constexpr bool value = true;  };
struct FalseT { static constexpr bool value = false; };

__global__ __launch_bounds__(256) void patchnet_wmma_f32(
    const float* __restrict__ x,    // (32,224,224,3)
    const float* __restrict__ ind,  // (32,729,10)
    float* __restrict__ out)        // (32,10,32,32,3)
{
  // A matrix, transposed + padded: As[(h*28+w)*16 + m] = ind[b, h*27+w, m]
  __shared__ float As[KPAD * 16];   // 48384 bytes

  const int tid = threadIdx.x;
  const int b   = blockIdx.x / 24;   // batch (scalar)
  const int grp = blockIdx.x % 24;   // 8 tiles per group, 24 groups = 192 tiles

  // ---- stage A into LDS (zeros at m>=10, w==27) ----
  const float* indb = ind + b * (GRID_HW * GRID_HW * KDIM);
  for (int idx = tid; idx < KPAD * 16; idx += 256) {
    const int kcol = idx >> 4;
    const int mm   = idx & 15;
    const int h    = kcol / WPAD;
    const int w    = kcol - h * WPAD;
    float v = 0.0f;
    if (mm < KDIM && w < GRID_HW) v = indb[(h * GRID_HW + w) * KDIM + mm];
    As[idx] = v;
  }
  __syncthreads();

  const int wave = __builtin_amdgcn_readfirstlane(tid) >> 5;  // scalar wave id
  const int lane = tid & 31;
  const int tile = grp * 8 + wave;   // 0..191
  const int i    = tile / 6;         // patch row 0..31 (96 cols per i = 6 tiles)
  const int m0   = (tile % 6) * 16;  // base (j*3+c) column of this tile

  const int N   = lane & 15;         // output column within tile
  const int sel = (lane >> 4) * 2;   // this half-wave holds B rows {sel, sel+1}
  const int m   = m0 + N;            // flattened (j,c)
  const int jl  = m / 3;             // patch col j for this lane

  // validity: 0 <= w*8 + jl - 12 <= 223  and  0 <= h*8 + i - 12 <= 223
  const int wlo = (jl >= 12) ? 0 : ((12 - jl + 7) >> 3);
  int whi = (235 - jl) >> 3; if (whi > 26) whi = 26;
  const int hlo = (i >= 12) ? 0 : ((12 - i + 7) >> 3);   // scalar
  int hhi = (235 - i) >> 3; if (hhi > 26) hhi = 26;      // scalar

  const float* xb = x + (size_t)b * SLICE;

  // Full tile GEMM over valid h bodies, double-buffered.
  auto tileGemm = [&](auto edge_tag) -> v8f {
    constexpr bool EDGE = decltype(edge_tag)::value;

    auto loadBody = [&](int hb, int rowb, v2f* av, float (*bv)[2]) {
      const int abase = hb * (WPAD * 16) + sel * 16 + N;   // LDS float index
      const int sb0   = rowb + sel * 24;                   // stream-0 voffset
#pragma unroll
      for (int s = 0; s < NSLOT; ++s) {
        av[s].x = As[abase + s * 64];
        av[s].y = As[abase + s * 64 + 16];
        const int  w0 = sel + 4 * s;        // compile-time per slot/half-wave
        const int  w1 = w0 + 1;
        const bool p0 = (w0 >= wlo) & (w0 <= whi);  // loop-invariant lane mask
        const bool p1 = (w1 >= wlo) & (w1 <= whi);
        int t0 = sb0 + 96 * s;              // in [-36, SLICE+35] (y valid here)
        int t1 = t0 + 24;
        if (EDGE) {                          // allocation edge: clamp (v_med3)
          t0 = t0 < 0 ? 0 : (t0 > SLICE - 1 ? SLICE - 1 : t0);
          t1 = t1 < 0 ? 0 : (t1 > SLICE - 1 ? SLICE - 1 : t1);
        }
        const float v0 = xb[t0];             // global_load_b32 (scale_offset)
        const float v1 = xb[t1];
        bv[s][0] = p0 ? v0 : 0.0f;           // v_cndmask w/ hoisted SGPR mask
        bv[s][1] = p1 ? v1 : 0.0f;
      }
    };

    v8f acc = {};
    auto mmaBody = [&](const v2f* av, const float (*bv)[2]) {
#pragma unroll
      for (int s = 0; s < NSLOT; ++s) {
        v2f bf;
        bf.x = bv[s][0];
        bf.y = bv[s][1];
        acc = __builtin_amdgcn_wmma_f32_16x16x4_f32(
            false, av[s], false, bf, (short)0, acc, false, false);
      }
    };

    // bodies hlo..hhi only (rows outside contribute zero -> skipped entirely)
    int hb   = hlo;
    int rowb = (hb * 8 + i - 12) * ROWELEM + m - 36;   // y valid from here on
    v2f   aA[NSLOT], aB[NSLOT];
    float bA[NSLOT][2], bB[NSLOT][2];

    loadBody(hb, rowb, aA, bA);
    ++hb;
    for (; hb + 1 <= hhi; hb += 2) {
      rowb += 8 * ROWELEM;
      loadBody(hb, rowb, aB, bB);        // overlaps the WMMAs below
      mmaBody(aA, bA);
      rowb += 8 * ROWELEM;
      loadBody(hb + 1, rowb, aA, bA);
      mmaBody(aB, bB);
    }
    if (hb <= hhi) {                     // odd remainder body
      rowb += 8 * ROWELEM;
      loadBody(hb, rowb, aB, bB);
      mmaBody(aA, bA);
      mmaBody(aB, bB);
    } else {
      mmaBody(aA, bA);
    }
    return acc;
  };

  v8f acc;
  if (b == 0 || b == 31) acc = tileGemm(TrueT{});   // uniform s_cbranch
  else                   acc = tileGemm(FalseT{});

  // C/D layout: lane&15 = N ; VGPR v holds row M = v + 8*(lane>>4)
  float* ob = out + (size_t)b * (KDIM * OUTK) + i * (PATCH * CHN) + m0 + N;
  const int kbase = (lane >> 4) * 8;
#pragma unroll
  for (int v = 0; v < 8; ++v) {
    const int k = kbase + v;
    if (k < KDIM) ob[k * OUTK] = acc[v];
  }
}

extern "C" void kernel_launch(void* const* d_in, const int* in_sizes, int n_in,
                              void* d_out, int out_size, void* d_ws, size_t ws_size,
                              hipStream_t stream) {
  (void)in_sizes; (void)n_in; (void)d_ws; (void)ws_size; (void)out_size;
  const float* x   = (const float*)d_in[0];   // (32,224,224,3) f32
  const float* ind = (const float*)d_in[1];   // (32,729,10)    f32
  float* out = (float*)d_out;                 // (32,10,32,32,3) f32

  dim3 grid(32 * 24);   // batch * tile-group  (6144 16x16 output tiles total)
  dim3 block(256);      // 8 wave32 waves, one tile each
  patchnet_wmma_f32<<<grid, block, 0, stream>>>(x, ind, out);
}